// EmbeddingDropout_7318624272856
// MI455X (gfx1250) — compile-verified
//
#include <hip/hip_runtime.h>
#include <stdint.h>

// ---------------------------------------------------------------------------
// EmbeddingDropout for MI455X (gfx1250).
// out[tok, :] = mask(words[tok]) * weight[words[tok], :]
// VOCAB=50257, DIM=1024, TOKENS = 8*2048 = 16384, p_drop = 0.1
//
// Pure bandwidth problem (~121 MB HBM traffic -> ~5 us at 23.3 TB/s).
// Kernels:
//   1) mask_kernel: Threefry-2x32 per vocab row -> d_ws (201 KB), runs once.
//   2) embed_gather_kernel: async global->LDS B128 row copies (ASYNCcnt
//      pipeline), s_wait_asynccnt in-order drain, non-temporal B128 stores.
//   3) embed_fused_kernel (fallback, ws too small): per-block masks computed
//      by 8 VALU lanes -> LDS -> broadcast (not per-wave scalar recompute).
// ---------------------------------------------------------------------------

typedef __attribute__((ext_vector_type(4))) float v4f;
typedef __attribute__((ext_vector_type(4))) int   v4i;

typedef __attribute__((address_space(1))) v4i* gv4i_ptr;
typedef __attribute__((address_space(3))) v4i* lv4i_ptr;

#define VOCAB   50257
#define DIM     1024
#define TPB     8            // tokens per block
#define BLOCKT  256          // 8 waves (wave32)

#if __has_builtin(__builtin_amdgcn_global_load_async_to_lds_b128) && \
    __has_builtin(__builtin_amdgcn_s_wait_asynccnt)
#define HAVE_ASYNC 1
#else
#define HAVE_ASYNC 0
#endif

__device__ __forceinline__ uint32_t rotl32(uint32_t x, int n) {
    return (x << n) | (x >> (32 - n));
}

// Exact replica of jax.random.bernoulli(jax.random.key(42), 0.9, (VOCAB,1))
// for row v, returning keep ? 1/0.9 : 0.
// Threefry-2x32, key = {0, 42}. random_bits pads 50257 -> 50258 and splits
// into halves of H = 25129: pair i = (iota[i], iota[H+i]) with iota[50257]=0.
__device__ __forceinline__ float row_scale(uint32_t v) {
    const uint32_t H = (VOCAB + 1) / 2;   // 25129
    uint32_t x0, x1;
    const bool hi = (v >= H);
    if (hi) { x0 = v - H; x1 = v; }
    else    { x0 = v;     x1 = (v == H - 1) ? 0u : v + H; }

    const uint32_t ks0 = 0u;
    const uint32_t ks1 = 42u;
    const uint32_t ks2 = 0u ^ 42u ^ 0x1BD11BDAu;

    x0 += ks0; x1 += ks1;
#define TF_R(r) { x0 += x1; x1 = rotl32(x1, r); x1 ^= x0; }
    TF_R(13) TF_R(15) TF_R(26) TF_R(6)   x0 += ks1; x1 += ks2 + 1u;
    TF_R(17) TF_R(29) TF_R(16) TF_R(24)  x0 += ks2; x1 += ks0 + 2u;
    TF_R(13) TF_R(15) TF_R(26) TF_R(6)   x0 += ks0; x1 += ks1 + 3u;
    TF_R(17) TF_R(29) TF_R(16) TF_R(24)  x0 += ks1; x1 += ks2 + 4u;
    TF_R(13) TF_R(15) TF_R(26) TF_R(6)   x0 += ks2; x1 += ks0 + 5u;
#undef TF_R

    const uint32_t bits = hi ? x1 : x0;
    // uniform in [0,1): bitcast(bits>>9 | 0x3F800000) - 1.0f ; keep if < 0.9
    const float u = __uint_as_float((bits >> 9) | 0x3F800000u) - 1.0f;
    return (u < 0.9f) ? (1.0f / 0.9f) : 0.0f;
}

// ---- kernel 1: per-vocab-row dropout scale table (runs once, ~1571 waves) ----
__global__ __launch_bounds__(BLOCKT) void mask_kernel(float* __restrict__ mask) {
    const uint32_t v = blockIdx.x * BLOCKT + threadIdx.x;
    if (v < VOCAB) mask[v] = row_scale(v);
}

#if HAVE_ASYNC
template <int N>
__device__ __forceinline__ void wait_async() {
    __builtin_amdgcn_s_wait_asynccnt(N);
}

__device__ __forceinline__ void async_row_copy16(const float* gsrc, float* ldst) {
    // Per-lane: copies 16 bytes from the global address gsrc to the LDS address ldst (B128, tracked by ASYNCcnt)
    __builtin_amdgcn_global_load_async_to_lds_b128(
        (gv4i_ptr)(uintptr_t)gsrc,
        (lv4i_ptr)(uint32_t)(uintptr_t)ldst,
        0, 0);
}
#endif

// ---- kernel 2 (hot path): gather + scale, mask table from d_ws --------------
__global__ __launch_bounds__(BLOCKT) void embed_gather_kernel(
    const int* __restrict__ words, const float* __restrict__ weight,
    const float* __restrict__ mask, float* __restrict__ out)
{
    const int tid  = threadIdx.x;          // 0..255, each owns 4 floats of a row
    const int base = blockIdx.x * TPB;     // first token of this block
    const int col  = tid * 4;              // float offset within row

#if HAVE_ASYNC
    // 8 distinct 4KB buffers -> no LDS reuse, no WAR hazard, no barriers
    // (each lane reads back exactly the 16B its own async lane wrote).
    __shared__ float lbuf[TPB][DIM];       // 32 KB of 320 KB

    float scl[TPB];
    #pragma unroll
    for (int t = 0; t < TPB; ++t) {
        const uint32_t w = (uint32_t)words[base + t];   // one s_load_b256 total
        async_row_copy16(weight + (size_t)w * DIM + col, &lbuf[t][col]);
        scl[t] = mask[w];                  // uniform -> scalar load, hidden
    }

    // Async loads complete in order: after 8 issues, asynccnt<=7-t <=> copy t done.
    #define STEP(t, wn)                                                        \
    {                                                                          \
        wait_async<wn>();                                                      \
        v4f v = *(const v4f*)&lbuf[t][col];                                    \
        v *= scl[t];                                                           \
        __builtin_nontemporal_store(                                           \
            v, (v4f*)(out + (size_t)(base + t) * DIM + col));                  \
    }
    STEP(0, 7); STEP(1, 6); STEP(2, 5); STEP(3, 4);
    STEP(4, 3); STEP(5, 2); STEP(6, 1); STEP(7, 0);
    #undef STEP
#else
    #pragma unroll
    for (int t = 0; t < TPB; ++t) {
        const uint32_t w = (uint32_t)words[base + t];
        const float s = mask[w];
        v4f v = *(const v4f*)(weight + (size_t)w * DIM + col);
        v *= s;
        __builtin_nontemporal_store(v, (v4f*)(out + (size_t)(base + t) * DIM + col));
    }
#endif
}

// ---- fallback (ws too small): masks computed by 8 VALU lanes, via LDS -------
__global__ __launch_bounds__(BLOCKT) void embed_fused_kernel(
    const int* __restrict__ words, const float* __restrict__ weight,
    float* __restrict__ out)
{
    const int tid  = threadIdx.x;
    const int base = blockIdx.x * TPB;
    const int col  = tid * 4;

    __shared__ float smask[TPB];
    if (tid < TPB) {
        // Lane t computes token t's Threefry mask: one divergent VALU body
        // per block instead of ~1030 SALU ops per wave.
        smask[tid] = row_scale((uint32_t)words[base + tid]);
    }
    __syncthreads();

    #pragma unroll
    for (int t = 0; t < TPB; ++t) {
        const uint32_t w = (uint32_t)words[base + t];
        const float s = smask[t];
        v4f v = *(const v4f*)(weight + (size_t)w * DIM + col);
        v *= s;
        __builtin_nontemporal_store(v, (v4f*)(out + (size_t)(base + t) * DIM + col));
    }
}

extern "C" void kernel_launch(void* const* d_in, const int* in_sizes, int n_in,
                              void* d_out, int out_size, void* d_ws, size_t ws_size,
                              hipStream_t stream) {
    const int*   words  = (const int*)d_in[0];    // [8, 2048] int32
    const float* weight = (const float*)d_in[1];  // [50257, 1024] f32
    float*       out    = (float*)d_out;          // [8, 2048, 1024] f32

    const int ntok   = in_sizes[0];               // 16384
    const int blocks = ntok / TPB;                // 2048

    if (ws_size >= (size_t)VOCAB * sizeof(float)) {
        float* mask = (float*)d_ws;
        mask_kernel<<<(VOCAB + BLOCKT - 1) / BLOCKT, BLOCKT, 0, stream>>>(mask);
        embed_gather_kernel<<<blocks, BLOCKT, 0, stream>>>(words, weight, mask, out);
    } else {
        embed_fused_kernel<<<blocks, BLOCKT, 0, stream>>>(words, weight, out);
    }
}